// Scenario_Robust_OPF_17652315586610
// MI455X (gfx1250) — compile-verified
//
#include <hip/hip_runtime.h>
#include <math.h>

typedef __attribute__((ext_vector_type(2))) float v2f;
typedef __attribute__((ext_vector_type(8))) float v8f;

#define NT 512  // 16 wave32 on one WGP: safe residency for a single persistent workgroup

constexpr int NG = 100, NL = 400, NN = 300, NWF = 50, NS = 200;
constexpr int G_P = 112;   // NG padded to 7*16
constexpr int S_P = 208;   // NS padded to 13*16
constexpr int PDHG_ITERS = 300, POWER_ITERS = 25;

// ---- x-space layout (padded; pads are zeroed once and never written) ----
constexpr int OX_P  = 0;
constexpr int OX_RU = OX_P  + G_P;
constexpr int OX_RD = OX_RU + G_P;
constexpr int OX_U  = OX_RD + G_P;
constexpr int OX_D  = OX_U  + G_P * S_P;
constexpr int OX_FU = OX_D  + G_P * S_P;
constexpr int OX_FD = OX_FU + NL;
constexpr int XSZ   = OX_FD + NL;

// ---- y-space layout ----
constexpr int OY_Y1 = 0;
constexpr int OY_Y2 = 16;
constexpr int OY_Y3 = OY_Y2 + NL;
constexpr int OY_Y4 = OY_Y3 + NL;
constexpr int OY_Z1 = OY_Y4 + S_P;
constexpr int OY_Z2 = OY_Z1 + G_P;
constexpr int OY_Z3 = OY_Z2 + G_P;
constexpr int OY_Z4 = OY_Z3 + G_P * S_P;
constexpr int OY_Z5 = OY_Z4 + G_P * S_P;
constexpr int OY_Z6 = OY_Z5 + NL * S_P;
constexpr int YSZ   = OY_Z6 + NL * S_P;

// ---- workspace layout (floats) ----
constexpr int WS_PG   = 0;                    // PG  (NL x G_P)
constexpr int WS_PGT  = WS_PG   + NL * G_P;   // PG^T (G_P x NL)
constexpr int WS_PW   = WS_PGT  + G_P * NL;   // pw (NL x S_P)
constexpr int WS_B2   = WS_PW   + NL * S_P;
constexpr int WS_B3   = WS_B2   + NL;
constexpr int WS_B4   = WS_B3   + NL;
constexpr int WS_B1   = WS_B4   + S_P;
constexpr int WS_M1   = WS_B1   + 16;         // node_W @ w_scen.T (NN x S_P)
constexpr int WS_NV   = WS_M1   + NN * S_P;   // node_W@w_exp - node_L@Pd (NN)
constexpr int WS_X    = WS_NV   + NN + 4;
constexpr int WS_XB   = WS_X    + XSZ;
constexpr int WS_ATY  = WS_XB   + XSZ;
constexpr int WS_Y    = WS_ATY  + XSZ;
constexpr int WS_AX   = WS_Y    + YSZ;
constexpr int WS_T    = WS_AX   + YSZ;        // t = PG^T @ (z5-z6)  (G_P x S_P)
constexpr int WS_FLOW = WS_T    + G_P * S_P;  // flow = PG @ (U-D)   (NL x S_P)
constexpr int WS_GP   = WS_FLOW + NL * S_P;   // PG @ p (NL)
constexpr int WS_COL  = WS_GP   + NL;         // colsum(U-D) (S_P)
constexpr int WS_S3   = WS_COL  + S_P;
constexpr int WS_S4   = WS_S3   + G_P;
constexpr int WS_S5   = WS_S4   + G_P;
constexpr int WS_S6   = WS_S5   + NL;
constexpr int WS_GTY  = WS_S6   + NL;         // PG^T @ (y2-y3) (G_P)
constexpr int WS_SUMP = WS_GTY  + G_P;
constexpr int WS_TOTAL= WS_SUMP + 16;

// ======================= prep kernels (grid-parallel) =======================

__global__ void k_zero(float* w) {
    for (int i = blockIdx.x * blockDim.x + threadIdx.x; i < WS_TOTAL;
         i += gridDim.x * blockDim.x) w[i] = 0.0f;
}

__global__ void k_prep1(float* w, const float* w_scen, const float* node_W,
                        const float* node_L, const float* Pd, const float* w_exp) {
    const int NM1 = NN * NS;
    const int total = NM1 + NN + NS + 1;
    for (int idx = blockIdx.x * blockDim.x + threadIdx.x; idx < total;
         idx += gridDim.x * blockDim.x) {
        if (idx < NM1) {                         // M1[n,s] = sum_w node_W[n,w]*w_scen[s,w]
            int n = idx / NS, s = idx % NS;
            float a = 0.f;
            for (int ww = 0; ww < NWF; ++ww) a += node_W[n*NWF + ww] * w_scen[s*NWF + ww];
            w[WS_M1 + n*S_P + s] = a;
        } else if (idx < NM1 + NN) {             // NV[n] = (node_W@w_exp - node_L@Pd)[n]
            int n = idx - NM1;
            float a = 0.f;
            for (int ww = 0; ww < NWF; ++ww) a += node_W[n*NWF + ww] * w_exp[ww];
            for (int m = 0; m < NN; ++m)     a -= node_L[n*NN + m] * Pd[m];
            w[WS_NV + n] = a;
        } else if (idx < NM1 + NN + NS) {        // b4[s] = -sum_w w_scen[s,w]
            int s = idx - NM1 - NN;
            float a = 0.f;
            for (int ww = 0; ww < NWF; ++ww) a += w_scen[s*NWF + ww];
            w[WS_B4 + s] = -a;
        } else {                                 // b1 = sum(Pd) - sum(w_exp)
            float a = 0.f;
            for (int m = 0; m < NN; ++m)  a += Pd[m];
            for (int ww = 0; ww < NWF; ++ww) a -= w_exp[ww];
            w[WS_B1] = a;
        }
    }
}

__global__ void k_prep2(float* w, const float* PTDF, const float* node_G,
                        const float* Line_cap) {
    const int A = NL * NG;           // PG
    const int B = A + NG * NL;       // PGT
    const int C = B + NL * NS;       // PW
    const int D2 = C + NL;           // b2
    const int E = D2 + NL;           // b3
    for (int idx = blockIdx.x * blockDim.x + threadIdx.x; idx < E;
         idx += gridDim.x * blockDim.x) {
        if (idx < A) {
            int l = idx / NG, g = idx % NG;
            float a = 0.f;
            for (int n = 0; n < NN; ++n) a += PTDF[l*NN + n] * node_G[n*NG + g];
            w[WS_PG + l*G_P + g] = a;
        } else if (idx < B) {
            int r = idx - A; int g = r / NL, l = r % NL;
            float a = 0.f;
            for (int n = 0; n < NN; ++n) a += PTDF[l*NN + n] * node_G[n*NG + g];
            w[WS_PGT + g*NL + l] = a;
        } else if (idx < C) {
            int r = idx - B; int l = r / NS, s = r % NS;
            float a = 0.f;
            for (int n = 0; n < NN; ++n) a += PTDF[l*NN + n] * w[WS_M1 + n*S_P + s];
            w[WS_PW + l*S_P + s] = a;
        } else if (idx < D2) {
            int l = idx - C;
            float a = 0.f;
            for (int n = 0; n < NN; ++n) a += PTDF[l*NN + n] * w[WS_NV + n];
            w[WS_B2 + l] = Line_cap[l] - a;
        } else {
            int l = idx - D2;
            float a = 0.f;
            for (int n = 0; n < NN; ++n) a += PTDF[l*NN + n] * w[WS_NV + n];
            w[WS_B3 + l] = Line_cap[l] + a;
        }
    }
}

__global__ void k_setv(float* w) {  // v = ones in real x-slots (pads stay 0)
    const int total = 3*NG + 2*NG*NS + 2*NL;   // 41100
    float* X = w + WS_X;
    for (int idx = blockIdx.x * blockDim.x + threadIdx.x; idx < total;
         idx += gridDim.x * blockDim.x) {
        int off;
        if (idx < NG)                off = OX_P  + idx;
        else if (idx < 2*NG)         off = OX_RU + (idx - NG);
        else if (idx < 3*NG)         off = OX_RD + (idx - 2*NG);
        else if (idx < 3*NG + NG*NS) { int r = idx - 3*NG; off = OX_U + (r/NS)*S_P + (r%NS); }
        else if (idx < 3*NG + 2*NG*NS) { int r = idx - 3*NG - NG*NS; off = OX_D + (r/NS)*S_P + (r%NS); }
        else if (idx < 3*NG + 2*NG*NS + NL) off = OX_FU + (idx - 3*NG - 2*NG*NS);
        else                         off = OX_FD + (idx - 3*NG - 2*NG*NS - NL);
        X[off] = 1.0f;
    }
}

// ======================= solver device helpers =======================

__device__ __forceinline__ float block_reduce(float v, float* red, int tid) {
    red[tid] = v; __syncthreads();
    for (int off = NT >> 1; off > 0; off >>= 1) {
        if (tid < off) red[tid] += red[tid + off];
        __syncthreads();
    }
    float r = red[0]; __syncthreads();
    return r;
}

// A @ x : xb -> yb (raw, no b / no step)
__device__ void matvec_core(float* w, const float* xb, float* yb, int tid) {
    const int lane = tid & 31, wave = tid >> 5, nw = NT / 32;
    const float* PG = w + WS_PG;
    float* flow = w + WS_FLOW;
    __syncthreads();
    // ---- WMMA GEMM: flow(NL x S_P) = PG(NL x NG) @ (U-D)(NG x S_P), K=100 ----
    {
        const float* Ux = xb + OX_U; const float* Dx = xb + OX_D;
        const int NTile = 25 * 13;
        const int m16 = lane & 15;
        const int kh  = (lane >> 4) * 2;
        for (int tile = wave; tile < NTile; tile += nw) {
            int tm = tile / 13, tn = tile % 13;
            int m = tm * 16 + m16;
            int n = tn * 16 + m16;
            v8f acc = {0.f,0.f,0.f,0.f,0.f,0.f,0.f,0.f};
            const float* arow = PG + m * G_P;
            for (int k0 = 0; k0 < NG; k0 += 4) {
                int ka = k0 + kh;
                v2f a, b;
                a[0] = arow[ka]; a[1] = arow[ka + 1];
                int bi = ka * S_P + n;
                b[0] = Ux[bi]        - Dx[bi];
                b[1] = Ux[bi + S_P]  - Dx[bi + S_P];
                acc = __builtin_amdgcn_wmma_f32_16x16x4_f32(false, a, false, b,
                                                            (short)0, acc, false, false);
            }
            int mr = tm * 16 + ((lane >> 4) << 3);
            for (int i = 0; i < 8; ++i) flow[(mr + i) * S_P + n] = acc[i];
        }
    }
    // ---- small matvecs / reductions (fixed order, deterministic) ----
    if (tid < NL) {                                   // gp[l] = (PG @ p)[l]
        float s = 0.f; const float* r = PG + tid * G_P; const float* p = xb + OX_P;
        for (int g = 0; g < NG; ++g) s += r[g] * p[g];
        w[WS_GP + tid] = s;
    }
    if (tid < NS) {                                   // col[s] = sum_g (U-D)[g,s]
        float s = 0.f;
        const float* Ux = xb + OX_U + tid; const float* Dx = xb + OX_D + tid;
        for (int g = 0; g < NG; ++g) s += Ux[g * S_P] - Dx[g * S_P];
        w[WS_COL + tid] = s;
    }
    if (tid == NT - 1) {                              // sum(p)
        float s = 0.f;
        for (int g = 0; g < NG; ++g) s += xb[OX_P + g];
        w[WS_SUMP] = s;
    }
    __syncthreads();
    // ---- write y = A x ----
    if (tid == 0) yb[OY_Y1] = w[WS_SUMP];
    for (int l = tid; l < NL; l += NT) {
        float g = w[WS_GP + l];
        yb[OY_Y2 + l] =  g + xb[OX_FU + l];
        yb[OY_Y3 + l] = -g + xb[OX_FD + l];
    }
    for (int s = tid; s < NS; s += NT) yb[OY_Y4 + s] = w[WS_COL + s];
    for (int g = tid; g < NG; g += NT) {
        yb[OY_Z1 + g] = xb[OX_P + g]  + xb[OX_RU + g];
        yb[OY_Z2 + g] = xb[OX_RD + g] - xb[OX_P + g];
    }
    for (int i = tid; i < NG * NS; i += NT) {
        int g = i / NS, s = i % NS, o = g * S_P + s;
        yb[OY_Z3 + o] = xb[OX_U + o] - xb[OX_RU + g];
        yb[OY_Z4 + o] = xb[OX_D + o] - xb[OX_RD + g];
    }
    for (int i = tid; i < NL * NS; i += NT) {
        int l = i / NS, s = i % NS, o = l * S_P + s;
        float f = flow[o];
        yb[OY_Z5 + o] =  f - xb[OX_FU + l];
        yb[OY_Z6 + o] = -f - xb[OX_FD + l];
    }
}

// A^T @ y : yb -> xo (raw)
__device__ void rmatvec_core(float* w, const float* yb, float* xo, int tid) {
    const int lane = tid & 31, wave = tid >> 5, nw = NT / 32;
    const float* PGT = w + WS_PGT;
    float* T = w + WS_T;
    __syncthreads();
    // ---- WMMA GEMM: T(G_P x S_P) = PGT(G_P x NL) @ (z5-z6)(NL x S_P), K=400 ----
    {
        const float* Z5 = yb + OY_Z5; const float* Z6 = yb + OY_Z6;
        const int NTile = 7 * 13;
        const int m16 = lane & 15;
        const int kh  = (lane >> 4) * 2;
        for (int tile = wave; tile < NTile; tile += nw) {
            int tm = tile / 13, tn = tile % 13;
            int m = tm * 16 + m16;
            int n = tn * 16 + m16;
            v8f acc = {0.f,0.f,0.f,0.f,0.f,0.f,0.f,0.f};
            const float* arow = PGT + m * NL;
            for (int k0 = 0; k0 < NL; k0 += 4) {
                int ka = k0 + kh;
                v2f a, b;
                a[0] = arow[ka]; a[1] = arow[ka + 1];
                int bi = ka * S_P + n;
                b[0] = Z5[bi]       - Z6[bi];
                b[1] = Z5[bi + S_P] - Z6[bi + S_P];
                acc = __builtin_amdgcn_wmma_f32_16x16x4_f32(false, a, false, b,
                                                            (short)0, acc, false, false);
            }
            int mr = tm * 16 + ((lane >> 4) << 3);
            for (int i = 0; i < 8; ++i) T[(mr + i) * S_P + n] = acc[i];
        }
    }
    // ---- row sums + small transposed matvec ----
    if (tid < NG) {
        float a = 0.f, b = 0.f, c = 0.f;
        const float* z3r = yb + OY_Z3 + tid * S_P;
        const float* z4r = yb + OY_Z4 + tid * S_P;
        for (int s = 0; s < NS; ++s) { a += z3r[s]; b += z4r[s]; }
        const float* pr = PGT + tid * NL;
        for (int l = 0; l < NL; ++l) c += pr[l] * (yb[OY_Y2 + l] - yb[OY_Y3 + l]);
        w[WS_S3 + tid] = a; w[WS_S4 + tid] = b; w[WS_GTY + tid] = c;
    }
    if (tid >= NG && tid < NG + NL) {
        int l = tid - NG;
        float a = 0.f, b = 0.f;
        const float* z5r = yb + OY_Z5 + l * S_P;
        const float* z6r = yb + OY_Z6 + l * S_P;
        for (int s = 0; s < NS; ++s) { a += z5r[s]; b += z6r[s]; }
        w[WS_S5 + l] = a; w[WS_S6 + l] = b;
    }
    __syncthreads();
    // ---- write x = A^T y ----
    float y1v = yb[OY_Y1];
    for (int g = tid; g < NG; g += NT) {
        xo[OX_P  + g] = y1v + w[WS_GTY + g] + yb[OY_Z1 + g] - yb[OY_Z2 + g];
        xo[OX_RU + g] = yb[OY_Z1 + g] - w[WS_S3 + g];
        xo[OX_RD + g] = yb[OY_Z2 + g] - w[WS_S4 + g];
    }
    for (int l = tid; l < NL; l += NT) {
        xo[OX_FU + l] = yb[OY_Y2 + l] - w[WS_S5 + l];
        xo[OX_FD + l] = yb[OY_Y3 + l] - w[WS_S6 + l];
    }
    for (int i = tid; i < NG * NS; i += NT) {
        int g = i / NS, s = i % NS, o = g * S_P + s;
        float tv = T[o];
        xo[OX_U + o] =  yb[OY_Y4 + s] + yb[OY_Z3 + o] + tv;
        xo[OX_D + o] = -yb[OY_Y4 + s] + yb[OY_Z4 + o] - tv;
    }
}

// ======================= persistent single-WGP solver =======================

__global__ __launch_bounds__(NT) void k_solver(float* w, const float* Cost,
                                               const float* Cru, const float* Crd,
                                               const float* Pmax, float* out) {
    __shared__ float red[NT];
    const int tid = threadIdx.x;
    float* X   = w + WS_X;
    float* XB  = w + WS_XB;
    float* ATY = w + WS_ATY;
    float* Y   = w + WS_Y;
    float* AX  = w + WS_AX;

    // warm PG / PGT into cache (gfx1250 global_prefetch_b8)
    for (int i = tid * 32; i < NL * G_P; i += NT * 32) {
        __builtin_prefetch(&w[WS_PG + i], 0, 3);
        __builtin_prefetch(&w[WS_PGT + i], 0, 3);
    }

    // ---- power iteration for ||A|| (v lives in X, initialized to ones) ----
    float sumsq = 0.f;
    for (int it = 0; it < POWER_ITERS; ++it) {
        __syncthreads();
        float part = 0.f;
        for (int i = tid; i < XSZ; i += NT) { float v = X[i]; part += v * v; }
        sumsq = block_reduce(part, red, tid);
        float inv = 1.0f / sqrtf(sumsq);
        for (int i = tid; i < XSZ; i += NT) X[i] *= inv;
        matvec_core(w, X, AX, tid);
        rmatvec_core(w, AX, X, tid);
    }
    __syncthreads();
    {
        float part = 0.f;
        for (int i = tid; i < XSZ; i += NT) { float v = X[i]; part += v * v; }
        sumsq = block_reduce(part, red, tid);
    }
    const float step = 0.9f / sqrtf(sqrtf(sumsq));   // 0.9 / sqrt(_tree_norm(v))
    const float b1 = w[WS_B1];
    const float* pw = w + WS_PW;

    // ---- reset state for PDHG ----
    for (int i = tid; i < XSZ; i += NT) { X[i] = 0.f; XB[i] = 0.f; }
    for (int i = tid; i < YSZ; i += NT) Y[i] = 0.f;

    // ---- PDHG loop ----
    for (int it = 0; it < PDHG_ITERS; ++it) {
        rmatvec_core(w, Y, ATY, tid);               // aty = A^T y (internal barriers)
        __syncthreads();
        // x_{n+1} = max(x - step*(c + aty), 0);  xb = 2 x_{n+1} - x
        for (int g = tid; g < NG; g += NT) {
            float o, n;
            o = X[OX_P  + g]; n = fmaxf(o - step * (Cost[g] + ATY[OX_P  + g]), 0.f);
            X[OX_P  + g] = n; XB[OX_P  + g] = 2.f * n - o;
            o = X[OX_RU + g]; n = fmaxf(o - step * (Cru[g]  + ATY[OX_RU + g]), 0.f);
            X[OX_RU + g] = n; XB[OX_RU + g] = 2.f * n - o;
            o = X[OX_RD + g]; n = fmaxf(o - step * (Crd[g]  + ATY[OX_RD + g]), 0.f);
            X[OX_RD + g] = n; XB[OX_RD + g] = 2.f * n - o;
        }
        for (int l = tid; l < NL; l += NT) {
            float o, n;
            o = X[OX_FU + l]; n = fmaxf(o - step * ATY[OX_FU + l], 0.f);
            X[OX_FU + l] = n; XB[OX_FU + l] = 2.f * n - o;
            o = X[OX_FD + l]; n = fmaxf(o - step * ATY[OX_FD + l], 0.f);
            X[OX_FD + l] = n; XB[OX_FD + l] = 2.f * n - o;
        }
        for (int i = tid; i < NG * NS; i += NT) {
            int g = i / NS, s = i % NS, o0 = g * S_P + s;
            float o, n;
            o = X[OX_U + o0]; n = fmaxf(o - step * ATY[OX_U + o0], 0.f);
            X[OX_U + o0] = n; XB[OX_U + o0] = 2.f * n - o;
            o = X[OX_D + o0]; n = fmaxf(o - step * ATY[OX_D + o0], 0.f);
            X[OX_D + o0] = n; XB[OX_D + o0] = 2.f * n - o;
        }
        matvec_core(w, XB, AX, tid);                // ax = A (2x_{n+1} - x)
        __syncthreads();
        // y_{n+1} = y + step*(ax - b), clamp inequality duals
        if (tid == 0) Y[OY_Y1] += step * (AX[OY_Y1] - b1);
        for (int l = tid; l < NL; l += NT) {
            Y[OY_Y2 + l] += step * (AX[OY_Y2 + l] - w[WS_B2 + l]);
            Y[OY_Y3 + l] += step * (AX[OY_Y3 + l] - w[WS_B3 + l]);
        }
        for (int s = tid; s < NS; s += NT)
            Y[OY_Y4 + s] += step * (AX[OY_Y4 + s] - w[WS_B4 + s]);
        for (int g = tid; g < NG; g += NT) {
            Y[OY_Z1 + g] = fmaxf(Y[OY_Z1 + g] + step * (AX[OY_Z1 + g] - Pmax[g]), 0.f);
            Y[OY_Z2 + g] = fmaxf(Y[OY_Z2 + g] + step *  AX[OY_Z2 + g], 0.f);
        }
        for (int i = tid; i < NG * NS; i += NT) {
            int g = i / NS, s = i % NS, o = g * S_P + s;
            Y[OY_Z3 + o] = fmaxf(Y[OY_Z3 + o] + step * AX[OY_Z3 + o], 0.f);
            Y[OY_Z4 + o] = fmaxf(Y[OY_Z4 + o] + step * AX[OY_Z4 + o], 0.f);
        }
        for (int i = tid; i < NL * NS; i += NT) {
            int l = i / NS, s = i % NS, o = l * S_P + s;
            Y[OY_Z5 + o] = fmaxf(Y[OY_Z5 + o] + step * (AX[OY_Z5 + o] + pw[o]), 0.f); // b9 = -pw
            Y[OY_Z6 + o] = fmaxf(Y[OY_Z6 + o] + step * (AX[OY_Z6 + o] - pw[o]), 0.f); // b10 = +pw
        }
        __syncthreads();
    }

    // ---- emit x = (p, r_up, r_down, U, D, f_up, f_down) un-padded ----
    for (int g = tid; g < NG; g += NT) {
        out[g]          = X[OX_P  + g];
        out[NG + g]     = X[OX_RU + g];
        out[2 * NG + g] = X[OX_RD + g];
    }
    for (int i = tid; i < NG * NS; i += NT) {
        int g = i / NS, s = i % NS;
        out[3 * NG + i]           = X[OX_U + g * S_P + s];
        out[3 * NG + NG * NS + i] = X[OX_D + g * S_P + s];
    }
    for (int l = tid; l < NL; l += NT) {
        out[3 * NG + 2 * NG * NS + l]      = X[OX_FU + l];
        out[3 * NG + 2 * NG * NS + NL + l] = X[OX_FD + l];
    }
}

// ======================= host entry =======================

extern "C" void kernel_launch(void* const* d_in, const int* in_sizes, int n_in,
                              void* d_out, int out_size, void* d_ws, size_t ws_size,
                              hipStream_t stream) {
    const float* w_scen   = (const float*)d_in[0];
    const float* Cost     = (const float*)d_in[1];
    const float* C_r_up   = (const float*)d_in[2];
    const float* C_r_down = (const float*)d_in[3];
    const float* Pmax     = (const float*)d_in[4];
    const float* PTDF     = (const float*)d_in[5];
    const float* node_G   = (const float*)d_in[6];
    const float* node_W   = (const float*)d_in[7];
    const float* node_L   = (const float*)d_in[8];
    const float* Pd       = (const float*)d_in[9];
    const float* w_exp    = (const float*)d_in[10];
    const float* Line_cap = (const float*)d_in[11];
    float* w   = (float*)d_ws;
    float* out = (float*)d_out;

    k_zero <<<512, 256, 0, stream>>>(w);
    k_prep1<<<240, 256, 0, stream>>>(w, w_scen, node_W, node_L, Pd, w_exp);
    k_prep2<<<640, 256, 0, stream>>>(w, PTDF, node_G, Line_cap);
    k_setv <<<161, 256, 0, stream>>>(w);
    k_solver<<<1, NT, 0, stream>>>(w, Cost, C_r_up, C_r_down, Pmax, out);
}